// FullLeNet_36283883717260
// MI455X (gfx1250) — compile-verified
//
#include <hip/hip_runtime.h>
#include <math.h>

typedef float v2f __attribute__((ext_vector_type(2)));
typedef float v8f __attribute__((ext_vector_type(8)));
typedef unsigned int v4u __attribute__((ext_vector_type(4)));
typedef int v4i __attribute__((ext_vector_type(4)));
typedef int v8i __attribute__((ext_vector_type(8)));

// V_WMMA_F32_16X16X4_F32: A 16x4 f32 (2 VGPR), B 4x16 f32 (2 VGPR), C/D 16x16 f32 (8 VGPR)
__device__ __forceinline__ v8f wmma4(v2f a, v2f b, v8f c) {
  return __builtin_amdgcn_wmma_f32_16x16x4_f32(
      /*neg_a=*/false, a, /*neg_b=*/false, b,
      /*c_mod=*/(short)0, c, /*reuse_a=*/false, /*reuse_b=*/false);
}

// ---------------------------------------------------------------------------
// TDM: 1-D tensor_load_to_lds of `nelem` f32 (nelem <= 65535) from global->LDS.
// D# per CDNA5 ISA ch.8: group0 = {count=1, lds_addr, global_addr[56:0], type=2}
// group1 = {wg_mask=0, data_size=4B, tensor_dim0=nelem, tensor_dim1=1,
//           tile_dim0=nelem, tile_dim1=0(unused), dim0_stride=nelem}
// Issue from ONE wave per block (EXEC ignored; instruction is per-wave).
// Tracked by TENSORcnt -> s_wait_tensorcnt 0 before the barrier.
// ---------------------------------------------------------------------------
__device__ __forceinline__ void tdm_load_1d_f32(void* lds_dst, const void* gsrc,
                                                unsigned nelem) {
  unsigned lds_off = (unsigned)(unsigned long long)lds_dst;  // flat LDS addr low 32b = offset
  unsigned long long ga = (unsigned long long)gsrc;
  v4u g0;
  g0.x = 1u;                                   // count=1 (valid user descriptor)
  g0.y = lds_off;                              // lds_addr (bytes)
  g0.z = (unsigned)(ga & 0xffffffffu);         // global_addr[31:0]
  g0.w = (unsigned)((ga >> 32) & 0x1ffffffu)   // global_addr[56:32]
         | (2u << 30);                         // type=2 ("image")
  v8i g1;
  g1[0] = (int)(2u << 16);                     // data_size=2 -> 4 bytes
  g1[1] = (int)((nelem & 0xffffu) << 16);      // tensor_dim0[15:0]  (bits 63:48)
  g1[2] = (int)(((nelem >> 16) & 0xffffu)      // tensor_dim0[31:16] (bits 79:64)
                | (1u << 16));                 // tensor_dim1 = 1    (bits 95:80)
  g1[3] = (int)((nelem & 0xffffu) << 16);      // tile_dim0 = nelem  (bits 127:112)
  g1[4] = 0;                                   // tile_dim1/2 = 0 (unused)
  g1[5] = (int)nelem;                          // tensor_dim0_stride[31:0]
  g1[6] = 0;                                   // stride[47:32]=0, dim1_stride lo=0
  g1[7] = 0;
  v4i z4 = {0, 0, 0, 0};
#if defined(__clang_major__) && __clang_major__ >= 23
  v8i z8 = {0, 0, 0, 0, 0, 0, 0, 0};
  __builtin_amdgcn_tensor_load_to_lds(g0, g1, z4, z4, z8, 0);
#else
  __builtin_amdgcn_tensor_load_to_lds(g0, g1, z4, z4, 0);
#endif
}

// ---------------------------------------------------------------------------
// Kernel 1: conv1 (1->16, 5x5, pad2) + bias + relu + 2x2 maxpool -> h1[B,16,14,14]
// Direct VALU fp32 (K=25 too skinny for WMMA). One block per image.
// ---------------------------------------------------------------------------
__global__ __launch_bounds__(128) void conv1_pool(
    const float* __restrict__ x, const float* __restrict__ w1,
    const float* __restrict__ b1, float* __restrict__ h1) {
  __shared__ float s_img[784];   // 28*28
  __shared__ float s_w[400];     // 16*25
  const int b = blockIdx.x, tid = threadIdx.x;
  if (tid < 32) tdm_load_1d_f32(s_img, x + (size_t)b * 784, 784);
  for (int i = tid; i < 400; i += 128) s_w[i] = w1[i];
  if (tid < 32) __builtin_amdgcn_s_wait_tensorcnt(0);
  __syncthreads();
  for (int i = tid; i < 3136; i += 128) {        // 16 ch * 14*14 pooled
    int oc = i / 196, r = i - oc * 196, py = r / 14, px = r - py * 14;
    const float* wk = &s_w[oc * 25];
    float bias = b1[oc];
    float best = -3.4e38f;
    for (int sy = 0; sy < 2; ++sy)
      for (int sx = 0; sx < 2; ++sx) {
        int oy = 2 * py + sy, ox = 2 * px + sx;
        float acc = bias;
        for (int ky = 0; ky < 5; ++ky) {
          int iy = oy + ky - 2; if (iy < 0 || iy >= 28) continue;
          for (int kx = 0; kx < 5; ++kx) {
            int ix = ox + kx - 2; if (ix < 0 || ix >= 28) continue;
            acc = fmaf(s_img[iy * 28 + ix], wk[ky * 5 + kx], acc);
          }
        }
        best = fmaxf(best, acc);
      }
    h1[(size_t)b * 3136 + i] = fmaxf(best, 0.f);   // relu(max) == max(relu)
  }
}

// ---------------------------------------------------------------------------
// Kernel 2: conv2 (16->32, 5x5, pad2) implicit-GEMM WMMA fp32 + relu + 2x2 pool
// M = 196 pixels -> 7 PAIRS of 16-row tiles per wave loop (B frags reused by
// both tiles of a pair), N = 32 (2 WMMA tiles), K = 400 (100 k4-steps).
// Stores are unguarded for full tiles (wave-uniform t2<6 test); only the one
// partial tile (rows 192..195) keeps a per-row guard.
// LDS layout: [wT | in | conv | kpk] so the +/-30-elem gather overshoot from
// `in` stays inside the allocation (value is cndmask'ed to 0, no exec branch).
// ---------------------------------------------------------------------------
__global__ __launch_bounds__(128) void conv2_pool(
    const float* __restrict__ h1, const float* __restrict__ w2,
    const float* __restrict__ b2, float* __restrict__ h2) {
  extern __shared__ float smem[];
  float* s_wT   = smem;                 // wT[k][oc] : 400*32 = 12800
  float* s_in   = s_wT + 12800;         // 16*14*14 = 3136
  float* s_conv = s_in + 3136;          // 208*32 = 6656
  int*   s_kpk  = (int*)(s_conv + 6656);// 400 packed entries
  const int b = blockIdx.x, tid = threadIdx.x;

  if (tid < 32) tdm_load_1d_f32(s_in, h1 + (size_t)b * 3136, 3136);
  for (int i = tid; i < 12800; i += 128) {
    int k = i >> 5, oc = i & 31;
    s_wT[i] = w2[oc * 400 + k];         // w2 is [32][16*25] flat
  }
  for (int k = tid; k < 400; k += 128) {
    int ic = k / 25, r = k - ic * 25, ky = r / 5, kx = r - ky * 5;
    int dy = ky - 2, dx = kx - 2;
    int koff = ic * 196 + dy * 14 + dx;          // gather offset rel. to pixel base
    s_kpk[k] = ((koff + 64) << 6) | ((dy + 2) << 3) | (dx + 2);
  }
  if (tid < 32) __builtin_amdgcn_s_wait_tensorcnt(0);
  __syncthreads();

  const int wid = tid >> 5, lane = tid & 31, half = lane >> 4, mrow = lane & 15;
  const float bias0 = b2[mrow], bias1 = b2[16 + mrow];

  for (int t2 = wid; t2 < 7; t2 += 4) {          // wave-uniform pair loop
    v8f c0a = {}, c1a = {}, c0b = {}, c1b = {};
    int pa = t2 * 32 + mrow, pb = pa + 16;
    int pa2 = (pa < 196) ? pa : 195;
    int pb2 = (pb < 196) ? pb : 195;
    int oyA = pa2 / 14, oxA = pa2 - oyA * 14, baseA = oyA * 14 + oxA;
    int oyB = pb2 / 14, oxB = pb2 - oyB * 14, baseB = oyB * 14 + oxB;
    for (int kk = 0; kk < 100; ++kk) {
      int k0 = kk * 4 + 2 * half, k1 = k0 + 1;   // A/B lane K = comp + 2*half
      int2 tt = *reinterpret_cast<const int2*>(s_kpk + k0);   // ds_load_b64
      int dx0 = (tt.x & 7) - 2, dy0 = ((tt.x >> 3) & 7) - 2, of0 = (tt.x >> 6) - 64;
      int dx1 = (tt.y & 7) - 2, dy1 = ((tt.y >> 3) & 7) - 2, of1 = (tt.y >> 6) - 64;
      float vA0 = s_in[baseA + of0], vA1 = s_in[baseA + of1];
      float vB0 = s_in[baseB + of0], vB1 = s_in[baseB + of1];
      bool okA0 = ((unsigned)(oyA + dy0) < 14u) & ((unsigned)(oxA + dx0) < 14u);
      bool okA1 = ((unsigned)(oyA + dy1) < 14u) & ((unsigned)(oxA + dx1) < 14u);
      bool okB0 = ((unsigned)(oyB + dy0) < 14u) & ((unsigned)(oxB + dx0) < 14u);
      bool okB1 = ((unsigned)(oyB + dy1) < 14u) & ((unsigned)(oxB + dx1) < 14u);
      v2f aA = {okA0 ? vA0 : 0.f, okA1 ? vA1 : 0.f};
      v2f aB = {okB0 ? vB0 : 0.f, okB1 ? vB1 : 0.f};
      v2f bb0 = {s_wT[k0 * 32 + mrow],      s_wT[k1 * 32 + mrow]};
      v2f bb1 = {s_wT[k0 * 32 + 16 + mrow], s_wT[k1 * 32 + 16 + mrow]};
      c0a = wmma4(aA, bb0, c0a);
      c1a = wmma4(aA, bb1, c1a);
      c0b = wmma4(aB, bb0, c0b);
      c1b = wmma4(aB, bb1, c1b);
    }
    int qa = t2 * 32 + 8 * half;                 // C layout: M = r + 8*half
    int qb = qa + 16;
    if (t2 < 6) {                                // tiles 0..11: always in-range
#pragma unroll
      for (int r = 0; r < 8; ++r) {
        s_conv[(qa + r) * 32 + mrow]      = fmaxf(c0a[r] + bias0, 0.f);
        s_conv[(qa + r) * 32 + 16 + mrow] = fmaxf(c1a[r] + bias1, 0.f);
        s_conv[(qb + r) * 32 + mrow]      = fmaxf(c0b[r] + bias0, 0.f);
        s_conv[(qb + r) * 32 + 16 + mrow] = fmaxf(c1b[r] + bias1, 0.f);
      }
    } else {                                     // tile 12 partial; tile 13 all OOB
#pragma unroll
      for (int r = 0; r < 8; ++r) {
        int q = qa + r;
        if (q < 196) {
          s_conv[q * 32 + mrow]      = fmaxf(c0a[r] + bias0, 0.f);
          s_conv[q * 32 + 16 + mrow] = fmaxf(c1a[r] + bias1, 0.f);
        }
      }
    }
  }
  __syncthreads();

  for (int i = tid; i < 1568; i += 128) {        // pool 14x14 -> 7x7, 32 ch
    int oc = i / 49, r = i - oc * 49, py = r / 7, px = r - py * 7;
    int y0 = 2 * py, x0 = 2 * px;
    float v = fmaxf(
        fmaxf(s_conv[(y0 * 14 + x0) * 32 + oc],       s_conv[(y0 * 14 + x0 + 1) * 32 + oc]),
        fmaxf(s_conv[((y0 + 1) * 14 + x0) * 32 + oc], s_conv[((y0 + 1) * 14 + x0 + 1) * 32 + oc]));
    h2[(size_t)b * 1568 + i] = v;
  }
}

// ---------------------------------------------------------------------------
// Kernel 3: conv3 (32->64, 5x5, pad2) implicit-GEMM WMMA fp32 + relu + pool pad1
// M = 49 pixels (4 tiles, 1/wave), N = 64 (4 WMMA tiles), K = 800 (200 k4-steps)
// ~231KB dynamic LDS (<= 320KB/WGP). Stores unguarded for waves 0..2. h3[B,1024]
// ---------------------------------------------------------------------------
__global__ __launch_bounds__(128) void conv3_pool(
    const float* __restrict__ h2, const float* __restrict__ w3,
    const float* __restrict__ b3, float* __restrict__ h3) {
  extern __shared__ float smem[];
  float* s_wT   = smem;                  // wT[k][oc] : 800*64 = 51200
  float* s_in   = s_wT + 51200;          // 32*7*7 = 1568
  float* s_conv = s_in + 1568;           // 64*64 = 4096
  int*   s_kpk  = (int*)(s_conv + 4096); // 800 packed entries
  const int b = blockIdx.x, tid = threadIdx.x;

  if (tid < 32) tdm_load_1d_f32(s_in, h2 + (size_t)b * 1568, 1568);
  for (int i = tid; i < 51200; i += 128) {
    int k = i >> 6, oc = i & 63;
    s_wT[i] = w3[oc * 800 + k];          // w3 is [64][32*25] flat
  }
  for (int k = tid; k < 800; k += 128) {
    int ic = k / 25, r = k - ic * 25, ky = r / 5, kx = r - ky * 5;
    int dy = ky - 2, dx = kx - 2;
    int koff = ic * 49 + dy * 7 + dx;
    s_kpk[k] = ((koff + 64) << 6) | ((dy + 2) << 3) | (dx + 2);
  }
  if (tid < 32) __builtin_amdgcn_s_wait_tensorcnt(0);
  __syncthreads();

  const int wid = tid >> 5, lane = tid & 31, half = lane >> 4, mrow = lane & 15;
  v8f c[4] = {};
  int p = wid * 16 + mrow;
  int p2 = (p < 49) ? p : 48;
  int oy = p2 / 7, ox = p2 - oy * 7;
  int pbase = oy * 7 + ox;
  for (int kk = 0; kk < 200; ++kk) {
    int k0 = kk * 4 + 2 * half, k1 = k0 + 1;
    int2 tt = *reinterpret_cast<const int2*>(s_kpk + k0);     // ds_load_b64
    int dx0 = (tt.x & 7) - 2, dy0 = ((tt.x >> 3) & 7) - 2, of0 = (tt.x >> 6) - 64;
    int dx1 = (tt.y & 7) - 2, dy1 = ((tt.y >> 3) & 7) - 2, of1 = (tt.y >> 6) - 64;
    float v0 = s_in[pbase + of0];
    float v1 = s_in[pbase + of1];
    bool ok0 = ((unsigned)(oy + dy0) < 7u) & ((unsigned)(ox + dx0) < 7u);
    bool ok1 = ((unsigned)(oy + dy1) < 7u) & ((unsigned)(ox + dx1) < 7u);
    v2f a = {ok0 ? v0 : 0.f, ok1 ? v1 : 0.f};
    const float* w0 = &s_wT[k0 * 64];
    const float* w1 = &s_wT[k1 * 64];
#pragma unroll
    for (int nt = 0; nt < 4; ++nt) {
      v2f bb = {w0[nt * 16 + mrow], w1[nt * 16 + mrow]};
      c[nt] = wmma4(a, bb, c[nt]);
    }
  }
  if (wid < 3) {                           // rows 0..47: always valid
#pragma unroll
    for (int r = 0; r < 8; ++r) {
      int q = wid * 16 + r + 8 * half;
#pragma unroll
      for (int nt = 0; nt < 4; ++nt) {
        int oc = nt * 16 + mrow;
        s_conv[q * 64 + oc] = fmaxf(c[nt][r] + b3[oc], 0.f);
      }
    }
  } else if (half == 0) {                  // wave 3: only row q=48 (r=0, half=0)
#pragma unroll
    for (int nt = 0; nt < 4; ++nt) {
      int oc = nt * 16 + mrow;
      s_conv[48 * 64 + oc] = fmaxf(c[nt][0] + b3[oc], 0.f);
    }
  }
  __syncthreads();

  // 2x2 maxpool, pad 1: 7x7 -> 4x4. s_conv is preceded by s_in in LDS, so the
  // y=-1/x=-1 offsets stay inside the allocation; select value, not address.
  for (int i = tid; i < 1024; i += 128) {
    int oc = i >> 4, r = i & 15, py = r >> 2, px = r & 3;
    float v = -3.4e38f;
#pragma unroll
    for (int dy = 0; dy < 2; ++dy) {
#pragma unroll
      for (int dx = 0; dx < 2; ++dx) {
        int y = 2 * py - 1 + dy, x = 2 * px - 1 + dx;
        bool ok = ((unsigned)y < 7u) & ((unsigned)x < 7u);
        float val = s_conv[(y * 7 + x) * 64 + oc];   // unconditional load
        v = fmaxf(v, ok ? val : -3.4e38f);           // v_cndmask
      }
    }
    h3[(size_t)b * 1024 + i] = v;   // matches reshape order [b, c*16+y*4+x]
  }
}

// ---------------------------------------------------------------------------
// Kernel 4: FC1  relu(h3[8192,1024] @ fw1^T + fb1) -> [8192,256], WMMA fp32
// Wave computes 16x64 tile; block(4 waves)=64x64; grid (M/64, 256/64)
// fw1 (1MB) lives in L2 (192MB) across all 512 blocks.
// ---------------------------------------------------------------------------
__global__ __launch_bounds__(128) void fc1_wmma(
    const float* __restrict__ h3, const float* __restrict__ fw1,
    const float* __restrict__ fb1, float* __restrict__ o) {
  const int tid = threadIdx.x, wid = tid >> 5, lane = tid & 31;
  const int half = lane >> 4, mrow = lane & 15;
  const int m0 = blockIdx.x * 64 + wid * 16;
  const int n0 = blockIdx.y * 64;
  v8f c[4] = {};
  const float* arow = h3 + (size_t)(m0 + mrow) * 1024;
  for (int kk = 0; kk < 256; ++kk) {
    int k0 = kk * 4 + 2 * half;
    v2f a = {arow[k0], arow[k0 + 1]};            // global_load_b64
#pragma unroll
    for (int nt = 0; nt < 4; ++nt) {
      const float* wr = fw1 + (size_t)(n0 + nt * 16 + mrow) * 1024;
      v2f bb = {wr[k0], wr[k0 + 1]};
      c[nt] = wmma4(a, bb, c[nt]);
    }
  }
  for (int r = 0; r < 8; ++r) {
    int row = m0 + r + 8 * half;
#pragma unroll
    for (int nt = 0; nt < 4; ++nt) {
      int col = n0 + nt * 16 + mrow;
      o[(size_t)row * 256 + col] = fmaxf(c[nt][r] + fb1[col], 0.f);
    }
  }
}

// ---------------------------------------------------------------------------
// Kernel 5: FC2 (256->10) + log_softmax. One wave32 per row.
// ---------------------------------------------------------------------------
__global__ __launch_bounds__(256) void fc2_lsm(
    const float* __restrict__ h, const float* __restrict__ fw2,
    const float* __restrict__ fb2, float* __restrict__ out, int B) {
  const int tid = threadIdx.x, wid = tid >> 5, lane = tid & 31;
  const int row = blockIdx.x * 8 + wid;
  if (row >= B) return;
  float acc[10];
#pragma unroll
  for (int j = 0; j < 10; ++j) acc[j] = 0.f;
  const float* hr = h + (size_t)row * 256;
  for (int i = 0; i < 8; ++i) {
    int k = lane + 32 * i;
    float hv = hr[k];
#pragma unroll
    for (int j = 0; j < 10; ++j) acc[j] = fmaf(hv, fw2[j * 256 + k], acc[j]);
  }
#pragma unroll
  for (int j = 0; j < 10; ++j)
    for (int off = 16; off > 0; off >>= 1)
      acc[j] += __shfl_xor(acc[j], off, 32);
  float z[10], m = -3.4e38f;
#pragma unroll
  for (int j = 0; j < 10; ++j) { z[j] = acc[j] + fb2[j]; m = fmaxf(m, z[j]); }
  float s = 0.f;
#pragma unroll
  for (int j = 0; j < 10; ++j) s += expf(z[j] - m);
  float lse = m + logf(s);
  if (lane < 10) out[(size_t)row * 10 + lane] = z[lane] - lse;
}

// ---------------------------------------------------------------------------
extern "C" void kernel_launch(void* const* d_in, const int* in_sizes, int n_in,
                              void* d_out, int out_size, void* d_ws, size_t ws_size,
                              hipStream_t stream) {
  const float* x   = (const float*)d_in[0];
  const float* w1  = (const float*)d_in[1];
  const float* b1  = (const float*)d_in[2];
  const float* w2  = (const float*)d_in[3];
  const float* b2  = (const float*)d_in[4];
  const float* w3  = (const float*)d_in[5];
  const float* b3  = (const float*)d_in[6];
  const float* fw1 = (const float*)d_in[7];
  const float* fb1 = (const float*)d_in[8];
  const float* fw2 = (const float*)d_in[9];
  const float* fb2 = (const float*)d_in[10];
  const int B = in_sizes[0] / 784;          // 8192

  float* h1 = (float*)d_ws;                 // [B,16,14,14]
  float* h2 = h1 + (size_t)B * 3136;        // [B,32,7,7]
  float* h3 = h2 + (size_t)B * 1568;        // [B,1024]
  float* f1 = h3 + (size_t)B * 1024;        // [B,256]

  conv1_pool<<<B, 128, 0, stream>>>(x, w1, b1, h1);

  size_t lds2 = (size_t)(12800 + 3136 + 6656) * sizeof(float) + 400 * sizeof(int);
  conv2_pool<<<B, 128, lds2, stream>>>(h1, w2, b2, h2);

  size_t lds3 = (size_t)(51200 + 1568 + 4096) * sizeof(float) + 800 * sizeof(int);
  conv3_pool<<<B, 128, lds3, stream>>>(h2, w3, b3, h3);

  dim3 g4(B / 64, 4);
  fc1_wmma<<<g4, 128, 0, stream>>>(h3, fw1, fb1, f1);

  fc2_lsm<<<(B + 7) / 8, 256, 0, stream>>>(f1, fw2, fb2, (float*)d_out, B);
}